// TemporalAttention_7756710937033
// MI455X (gfx1250) — compile-verified
//
#include <hip/hip_runtime.h>
#include <math.h>

// Problem constants (match reference)
#define BATCH  32
#define SEQ    2048
#define DMODEL 512     // D
#define DFFC   512     // DFF

typedef __attribute__((ext_vector_type(2))) float v2f;
typedef __attribute__((ext_vector_type(8))) float v8f;

// ---------------------------------------------------------------------------
// Workspace layout (floats):
//   [0,      16384)  xw    [B, D]      = sum_s Wt[s] * x[b,s,:]
//   [16384,  32768)  xta   [B, D]      = sum_s ta[b,s] * x[b,s,:]
//   [32768,  49152)  kw    [B, DFF]    = xw @ Wk^T + bk*Wtsum
//   [49152,  65536)  wq2   [B, D]      = kw @ Wq
//   [65536,  65568)  tasum [B]         = sum_s ta[b,s]
//   [65568]          wtsum             = sum_s Wt[s]
// ---------------------------------------------------------------------------
#define WS_XW     0
#define WS_XTA    16384
#define WS_KW     32768
#define WS_WQ2    49152
#define WS_TASUM  65536
#define WS_WTSUM  65568

// K0: zero the atomic accumulators, compute Wtsum. grid=129 blocks x 256.
__global__ void k0_init(float* __restrict__ ws, const float* __restrict__ wt) {
    int tid = threadIdx.x;
    int bid = blockIdx.x;
    if (bid < 128) {
        ws[bid * 256 + tid] = 0.0f;            // zeroes xw + xta (32768 floats)
    } else {
        if (tid < BATCH) ws[WS_TASUM + tid] = 0.0f;
        float p = 0.0f;
        for (int i = tid; i < SEQ; i += 256) p += wt[i];
        __shared__ float sred[256];
        sred[tid] = p;
        __syncthreads();
        for (int off = 128; off > 0; off >>= 1) {
            if (tid < off) sred[tid] += sred[tid + off];
            __syncthreads();
        }
        if (tid == 0) ws[WS_WTSUM] = sred[0];
    }
}

// K1: xw[b,d] = sum_s Wt[s] * x[b,s,d]. grid=(32, 2, 8), block=256.
// Coalesced across d; each block handles a 256-wide d chunk over 256 s rows.
__global__ void k1_xw(const float* __restrict__ x, const float* __restrict__ wt,
                      float* __restrict__ xw) {
    int b  = blockIdx.x;
    int d  = blockIdx.y * 256 + threadIdx.x;
    int s0 = blockIdx.z * 256;
    const float* xp = x + ((size_t)b * SEQ + s0) * DMODEL + d;
    const float* wp = wt + s0;
    float acc = 0.0f;
#pragma unroll 4
    for (int s = 0; s < 256; ++s)
        acc = fmaf(wp[s], xp[(size_t)s * DMODEL], acc);
    atomicAdd(&xw[b * DMODEL + d], acc);
}

// WMMA GEMM: Dout[m,n] = sum_k A[m,k]*B'[k,n] + bias[n]*rowscale[m*rsStride]
//   A: 32x512 row-major.  TRANS=1: B'[k,n] = W[n*512+k] (W is [N,K], e.g. Wk/Wv)
//                         TRANS=0: B'[k,n] = W[k*512+n] (W is [K,N], e.g. Wq)
// One wave per 16x16 output tile, V_WMMA_F32_16X16X4_F32 chained over K=512.
// A layout (ISA 7.12.2, 32-bit A 16x4): lanes 0-15 -> M=lane, K={k0,k0+1};
// lanes 16-31 -> M=lane-16, K={k0+2,k0+3}. B mirrors with N in place of M.
template <bool TRANS>
__global__ void k_gemm_wmma(const float* __restrict__ A, const float* __restrict__ W,
                            const float* __restrict__ bias,
                            const float* __restrict__ rowscale, int rsStride,
                            float* __restrict__ Dout) {
    const int K = DMODEL, N = DFFC;
    int lane = threadIdx.x;
    int l15  = lane & 15;
    int hi   = lane >> 4;            // 0 or 1
    int tm = blockIdx.x, tn = blockIdx.y;
    int mA = tm * 16 + l15;
    int nB = tn * 16 + l15;
    v8f acc = {};
    for (int k0 = 0; k0 < K; k0 += 4) {
        int ka = k0 + hi * 2;
        v2f a, bmat;
        a.x = A[mA * K + ka];
        a.y = A[mA * K + ka + 1];
        if (TRANS) {
            bmat.x = W[nB * K + ka];
            bmat.y = W[nB * K + ka + 1];
        } else {
            bmat.x = W[(size_t)ka * N + nB];
            bmat.y = W[(size_t)(ka + 1) * N + nB];
        }
        acc = __builtin_amdgcn_wmma_f32_16x16x4_f32(
            /*neg_a=*/false, a, /*neg_b=*/false, bmat,
            /*c_mod=*/(short)0, acc, /*reuse_a=*/false, /*reuse_b=*/false);
    }
    // C/D layout: lane<16 -> rows tm*16+0..7, lane>=16 -> rows tm*16+8..15; N=l15.
    float bn = bias ? bias[nB] : 0.0f;
    int mbase = tm * 16 + hi * 8;
#pragma unroll
    for (int g = 0; g < 8; ++g) {
        int m = mbase + g;
        float rs = rowscale ? rowscale[m * rsStride] : 0.0f;
        Dout[m * N + nB] = acc[g] + bn * rs;
    }
}

// K4: per (b,s): ta = (x[b,s,:]·wq2[b,:] + bq·kw[b,:]) / sqrt(DFF) + bt
//     accumulate xta[b,:] += ta*x[b,s,:],  tasum[b] += ta
// grid=(32, 8), block=256 (8 waves). Each wave owns 32 rows; wq2[b] cached
// in per-lane registers (4x float4); 512-wide dot via fma + shfl_xor allreduce.
__global__ void k4_ta(const float* __restrict__ x, const float* __restrict__ wq2,
                      const float* __restrict__ kw, const float* __restrict__ bq,
                      const float* __restrict__ bt, float* __restrict__ xta,
                      float* __restrict__ tasum) {
    const float INV = 0.04419417382415922f;  // 1/sqrt(512)
    int b   = blockIdx.x;
    int tid = threadIdx.x;
    int lane = tid & 31, wave = tid >> 5;

    __shared__ float sred[256];
    __shared__ float s_xta[DMODEL];
    __shared__ float s_ts;
    s_xta[tid]       = 0.0f;
    s_xta[tid + 256] = 0.0f;
    if (tid == 0) s_ts = 0.0f;

    // bq · kw[b]
    const float* kwb = kw + b * DFFC;
    sred[tid] = bq[tid] * kwb[tid] + bq[tid + 256] * kwb[tid + 256];
    __syncthreads();
    for (int off = 128; off > 0; off >>= 1) {
        if (tid < off) sred[tid] += sred[tid + off];
        __syncthreads();
    }
    float bq_kw = sred[0];
    float bt0   = bt[0];

    // wq2[b] resident in registers: lane covers d = c*128 + lane*4 .. +3
    const float4* wb = (const float4*)(wq2 + b * DMODEL);
    float4 w4[4];
#pragma unroll
    for (int c = 0; c < 4; ++c) w4[c] = wb[c * 32 + lane];

    float4 xacc[4];
#pragma unroll
    for (int c = 0; c < 4; ++c) xacc[c] = make_float4(0.f, 0.f, 0.f, 0.f);
    float tsacc = 0.0f;

    int srow0 = blockIdx.y * 256 + wave * 32;
    for (int r = 0; r < 32; ++r) {
        int s = srow0 + r;
        const float4* xr = (const float4*)(x + ((size_t)b * SEQ + s) * DMODEL);
        float4 x4[4];
        float dot = 0.0f;
#pragma unroll
        for (int c = 0; c < 4; ++c) {
            x4[c] = xr[c * 32 + lane];
            dot = fmaf(x4[c].x, w4[c].x, dot);
            dot = fmaf(x4[c].y, w4[c].y, dot);
            dot = fmaf(x4[c].z, w4[c].z, dot);
            dot = fmaf(x4[c].w, w4[c].w, dot);
        }
#pragma unroll
        for (int off = 16; off > 0; off >>= 1) dot += __shfl_xor(dot, off, 32);
        float ta = fmaf(dot + bq_kw, INV, bt0);
        tsacc += ta;
#pragma unroll
        for (int c = 0; c < 4; ++c) {
            xacc[c].x = fmaf(ta, x4[c].x, xacc[c].x);
            xacc[c].y = fmaf(ta, x4[c].y, xacc[c].y);
            xacc[c].z = fmaf(ta, x4[c].z, xacc[c].z);
            xacc[c].w = fmaf(ta, x4[c].w, xacc[c].w);
        }
    }
    // reduce across the 8 waves through LDS, then one pass of global atomics
#pragma unroll
    for (int c = 0; c < 4; ++c) {
        int dbase = c * 128 + lane * 4;
        atomicAdd(&s_xta[dbase + 0], xacc[c].x);
        atomicAdd(&s_xta[dbase + 1], xacc[c].y);
        atomicAdd(&s_xta[dbase + 2], xacc[c].z);
        atomicAdd(&s_xta[dbase + 3], xacc[c].w);
    }
    if (lane == 0) atomicAdd(&s_ts, tsacc);
    __syncthreads();
    atomicAdd(&xta[b * DMODEL + tid],       s_xta[tid]);
    atomicAdd(&xta[b * DMODEL + tid + 256], s_xta[tid + 256]);
    if (tid == 0) atomicAdd(&tasum[b], s_ts);
}

extern "C" void kernel_launch(void* const* d_in, const int* in_sizes, int n_in,
                              void* d_out, int out_size, void* d_ws, size_t ws_size,
                              hipStream_t stream) {
    const float* x  = (const float*)d_in[0];
    const float* Wq = (const float*)d_in[1];
    const float* bq = (const float*)d_in[2];
    const float* Wk = (const float*)d_in[3];
    const float* bk = (const float*)d_in[4];
    const float* Wv = (const float*)d_in[5];
    const float* bv = (const float*)d_in[6];
    const float* Wt = (const float*)d_in[7];
    const float* bt = (const float*)d_in[8];
    float* out = (float*)d_out;
    float* ws  = (float*)d_ws;

    float* xw    = ws + WS_XW;
    float* xta   = ws + WS_XTA;
    float* kw    = ws + WS_KW;
    float* wq2   = ws + WS_WQ2;
    float* tasum = ws + WS_TASUM;
    float* wtsum = ws + WS_WTSUM;

    // 0) zero accumulators + Wtsum
    k0_init<<<129, 256, 0, stream>>>(ws, Wt);
    // 1) xw[b,:] = sum_s Wt[s]*x[b,s,:]            (streams x once: 134 MB)
    k1_xw<<<dim3(BATCH, 2, 8), 256, 0, stream>>>(x, Wt, xw);
    // 2) kw = xw @ Wk^T + bk*Wtsum                 (WMMA, 32x512x512)
    k_gemm_wmma<true><<<dim3(2, 32), 32, 0, stream>>>(xw, Wk, bk, wtsum, 0, kw);
    // 3) wq2 = kw @ Wq                             (WMMA, 32x512x512)
    k_gemm_wmma<false><<<dim3(2, 32), 32, 0, stream>>>(kw, Wq, nullptr, nullptr, 0, wq2);
    // 4) ta, xta, tasum                            (streams x once more: 134 MB)
    k4_ta<<<dim3(BATCH, 8), 256, 0, stream>>>(x, wq2, kw, bq, bt, xta, tasum);
    // 5) latent = xta @ Wv^T + bv*tasum[b]         (WMMA, 32x512x512) -> d_out
    k_gemm_wmma<true><<<dim3(2, 32), 32, 0, stream>>>(xta, Wv, bv, tasum, 1, out);
}